// IterativeDecimator_78683800863309
// MI455X (gfx1250) — compile-verified
//
#include <hip/hip_runtime.h>
#include <hip/hip_bf16.h>

typedef __attribute__((ext_vector_type(2))) float v2f;
typedef __attribute__((ext_vector_type(8))) float v8f;

#define N_NODES 16384
#define LATENT  256
#define HIDDEN  32
#define KCLUST  64
#define N_EDGES 524288

// d_out layout (float element offsets), concatenated in return order:
#define OUT_COARSE_NODES 0        // 64*256 = 16384
#define OUT_CSEND  16384          // 4096
#define OUT_CRECV  20480          // 4096
#define OUT_CEDGE  24576          // 4096
#define OUT_ASSIGN 28672          // 16384*64 = 1048576

// workspace layout (byte offsets)
#define WS_CADJ_BYTES   ((size_t)0)                         // float[4096]
#define WS_KEEP_BYTES   ((size_t)16384)                     // u32[N_EDGES]
#define WS_BITMAP_BYTES ((size_t)16384 + (size_t)N_EDGES*4) // u32[N*N/32]
#define WS_MIN_BYTES    (WS_BITMAP_BYTES)
#define WS_DEDUP_BYTES  (WS_BITMAP_BYTES + (size_t)N_NODES*(size_t)N_NODES/8)

__device__ __forceinline__ v8f wmma_f32_k4(v2f a, v2f b, v8f c) {
  // D = A(16x4 f32) * B(4x16 f32) + C(16x16 f32); full-precision matrix pipe.
  return __builtin_amdgcn_wmma_f32_16x16x4_f32(false, a, false, b, (short)0, c,
                                               false, false);
}

// ---------------------------------------------------------------- kernel 0
__global__ void zero_kernel(float* __restrict__ cn, float* __restrict__ cadj,
                            unsigned* __restrict__ bitmap, long long bm_words) {
  long long i = (long long)blockIdx.x * blockDim.x + threadIdx.x;
  long long stride = (long long)gridDim.x * blockDim.x;
  for (long long j = i; j < 16384; j += stride) cn[j] = 0.0f;
  for (long long j = i; j < 4096;  j += stride) cadj[j] = 0.0f;
  for (long long j = i; j < bm_words; j += stride) bitmap[j] = 0u;
}

// ---------------------------------------------------------------- kernel 1
// Fused MLP (Dense32 -> relu -> Dense64) + softmax. One wave per 16-row tile,
// 4 waves / block. GEMM1: K=256 via 64 wmma steps x 2 N-tiles. h tile staged
// in LDS, GEMM2: K=32, 4 N-tiles. Softmax: 2 lanes/row + shfl_xor(16).
__global__ __launch_bounds__(128) void mlp_softmax_kernel(
    const float* __restrict__ X, const float* __restrict__ W1,
    const float* __restrict__ B1, const float* __restrict__ W2,
    const float* __restrict__ B2, float* __restrict__ S) {
  __shared__ float lds[4 * (16 * HIDDEN + 16 * KCLUST)];
  const int lane = threadIdx.x & 31;
  const int wave = threadIdx.x >> 5;
  const int half = lane >> 4;
  const int l16  = lane & 15;
  const int rowBase = (blockIdx.x * 4 + wave) * 16;
  float* hT = lds + wave * (16 * HIDDEN + 16 * KCLUST);
  float* lT = hT + 16 * HIDDEN;

  // ---- GEMM1: h = X @ W1  (accumulate two 16x16 N-tiles)
  v8f c0 = {}; v8f c1 = {};
  const float* arow = X + (size_t)(rowBase + l16) * LATENT + 2 * half;
  for (int k = 0; k < LATENT; k += 4) {
    v2f a = *(const v2f*)(arow + k);                    // A[m=l16][k..k+1 (+2 if half)]
    const float* w = W1 + (size_t)(k + 2 * half) * HIDDEN;
    v2f b0; b0.x = w[l16];      b0.y = w[HIDDEN + l16];
    v2f b1; b1.x = w[l16 + 16]; b1.y = w[HIDDEN + l16 + 16];
    c0 = wmma_f32_k4(a, b0, c0);
    c1 = wmma_f32_k4(a, b1, c1);
  }
  // bias + relu, stage h tile to LDS (row-major 16x32)
  {
    const float bias0 = B1[l16], bias1 = B1[16 + l16];
#pragma unroll
    for (int v = 0; v < 8; ++v) {
      const int m = v + 8 * half;                       // C layout: M = v + 8*half
      float h0 = c0[v] + bias0; h0 = h0 > 0.0f ? h0 : 0.0f;
      float h1 = c1[v] + bias1; h1 = h1 > 0.0f ? h1 : 0.0f;
      hT[m * HIDDEN + l16]      = h0;
      hT[m * HIDDEN + 16 + l16] = h1;
    }
  }
  __syncthreads();

  // ---- GEMM2: logits = h @ W2 (4 N-tiles of 16, K=32)
#pragma unroll
  for (int t = 0; t < 4; ++t) {
    v8f acc = {};
#pragma unroll
    for (int kk = 0; kk < HIDDEN; kk += 4) {
      v2f a; a.x = hT[l16 * HIDDEN + kk + 2 * half];
             a.y = hT[l16 * HIDDEN + kk + 1 + 2 * half];
      const float* w = W2 + (size_t)(kk + 2 * half) * KCLUST + t * 16;
      v2f b; b.x = w[l16]; b.y = w[KCLUST + l16];
      acc = wmma_f32_k4(a, b, acc);
    }
    const float bias = B2[t * 16 + l16];
#pragma unroll
    for (int v = 0; v < 8; ++v) {
      const int m = v + 8 * half;
      lT[m * KCLUST + t * 16 + l16] = acc[v] + bias;
    }
  }
  __syncthreads();

  // ---- softmax over 64 logits per row; lanes (l16, l16+16) pair on row l16
  {
    const float* lrow = lT + l16 * KCLUST + half * 32;
    float mx = -3.402823466e38f;
    for (int j = 0; j < 32; ++j) mx = fmaxf(mx, lrow[j]);
    mx = fmaxf(mx, __shfl_xor(mx, 16, 32));
    float sum = 0.0f;
    for (int j = 0; j < 32; ++j) sum += __expf(lrow[j] - mx);
    sum += __shfl_xor(sum, 16, 32);
    const float inv = 1.0f / sum;
    float* srow = S + (size_t)(rowBase + l16) * KCLUST + half * 32;
    for (int j = 0; j < 32; ++j) srow[j] = __expf(lrow[j] - mx) * inv;
  }
}

// ---------------------------------------------------------------- kernel 2
// Dedup: first claimant of bit (s*N+r) keeps the edge (matches .set(1.0)
// semantics; deterministic since duplicate edges contribute identically).
__global__ void keep_kernel(const int* __restrict__ snd, const int* __restrict__ rcv,
                            unsigned* __restrict__ keep, unsigned* __restrict__ bitmap) {
  const int e = blockIdx.x * blockDim.x + threadIdx.x;
  if (e >= N_EDGES) return;
  if (bitmap) {
    const unsigned idx = (unsigned)snd[e] * (unsigned)N_NODES + (unsigned)rcv[e];
    const unsigned old = atomicOr(&bitmap[idx >> 5], 1u << (idx & 31));
    keep[e] = ((old >> (idx & 31)) & 1u) ? 0u : 1u;
  } else {
    keep[e] = 1u;
  }
}

// ---------------------------------------------------------------- kernel 3
// coarse_nodes = S^T @ X : M=64, N=256, K=16384, K-split into 64 chunks of
// 256 nodes; one wave per (Mtile, Ntile, chunk); atomicAdd f32 reduction.
__global__ __launch_bounds__(128) void coarse_nodes_kernel(
    const float* __restrict__ S, const float* __restrict__ X, float* __restrict__ CN) {
  const int lane = threadIdx.x & 31;
  const int wave = threadIdx.x >> 5;
  const int half = lane >> 4, l16 = lane & 15;
  const int w = blockIdx.x * 4 + wave;   // 0..4095
  const int tile  = w & 63;              // 4 Mtiles x 16 Ntiles
  const int chunk = w >> 6;              // 0..63
  const int tm = tile >> 4, tn = tile & 15;
  const int nodeBase = chunk * 256;
  v8f acc = {};
  for (int kk = 0; kk < 256; kk += 4) {
    const int n0 = nodeBase + kk + 2 * half;
    v2f a; a.x = S[(size_t)n0 * KCLUST + tm * 16 + l16];       // A = S^T gather
           a.y = S[(size_t)(n0 + 1) * KCLUST + tm * 16 + l16];
    v2f b; b.x = X[(size_t)n0 * LATENT + tn * 16 + l16];
           b.y = X[(size_t)(n0 + 1) * LATENT + tn * 16 + l16];
    acc = wmma_f32_k4(a, b, acc);
  }
#pragma unroll
  for (int v = 0; v < 8; ++v) {
    const int m = tm * 16 + v + 8 * half;
    atomicAdd(&CN[(size_t)m * LATENT + tn * 16 + l16], acc[v]);
  }
}

// ---------------------------------------------------------------- kernel 4
// coarse_adj = S^T A S = sum over kept edges of outer(S[s], S[r]) expressed
// as a 64x64xE GEMM: A[:,e] = keep[e]*S[snd[e]], B[e,:] = S[rcv[e]].
// K-split into 256 chunks of 2048 edges; 16 MN tiles per chunk; gathers hit L2.
#define ADJ_CHUNK 2048
__global__ __launch_bounds__(128) void coarse_adj_kernel(
    const float* __restrict__ S, const int* __restrict__ snd,
    const int* __restrict__ rcv, const unsigned* __restrict__ keep,
    float* __restrict__ CADJ) {
  const int lane = threadIdx.x & 31;
  const int wave = threadIdx.x >> 5;
  const int half = lane >> 4, l16 = lane & 15;
  const int w = blockIdx.x * 4 + wave;   // 0..4095
  const int tile  = w & 15;              // 4x4 MN tiles
  const int chunk = w >> 4;              // 0..255
  const int tm = tile >> 2, tn = tile & 3;
  const int eBase = chunk * ADJ_CHUNK;
  v8f acc = {};
  for (int kk = 0; kk < ADJ_CHUNK; kk += 4) {
    const int e0 = eBase + kk + 2 * half;
    const int s0 = snd[e0], s1 = snd[e0 + 1];
    const int r0 = rcv[e0], r1 = rcv[e0 + 1];
    const float k0 = keep ? (float)keep[e0]     : 1.0f;
    const float k1 = keep ? (float)keep[e0 + 1] : 1.0f;
    v2f a; a.x = k0 * S[(size_t)s0 * KCLUST + tm * 16 + l16];
           a.y = k1 * S[(size_t)s1 * KCLUST + tm * 16 + l16];
    v2f b; b.x = S[(size_t)r0 * KCLUST + tn * 16 + l16];
           b.y = S[(size_t)r1 * KCLUST + tn * 16 + l16];
    acc = wmma_f32_k4(a, b, acc);
  }
#pragma unroll
  for (int v = 0; v < 8; ++v) {
    const int m = tm * 16 + v + 8 * half;
    atomicAdd(&CADJ[m * KCLUST + tn * 16 + l16], acc[v]);
  }
}

// ---------------------------------------------------------------- kernel 5
// jnp.nonzero(coarse_adj, size=K*K) row-major compaction; pad indices with 0
// so padded c_edges = coarse_adj[0,0]. 4096 entries -> serial is free.
__global__ void finalize_kernel(const float* __restrict__ CADJ, float* __restrict__ out) {
  if (threadIdx.x != 0 || blockIdx.x != 0) return;
  float* cs = out + OUT_CSEND;
  float* cr = out + OUT_CRECV;
  float* ce = out + OUT_CEDGE;
  int cnt = 0;
  for (int i = 0; i < KCLUST * KCLUST; ++i) {
    const float v = CADJ[i];
    if (v != 0.0f) { cs[cnt] = (float)(i >> 6); cr[cnt] = (float)(i & 63); ce[cnt] = v; ++cnt; }
  }
  const float fill = CADJ[0];
  for (int i = cnt; i < KCLUST * KCLUST; ++i) { cs[i] = 0.0f; cr[i] = 0.0f; ce[i] = fill; }
}

// ----------------------------------------------------------------- launcher
extern "C" void kernel_launch(void* const* d_in, const int* in_sizes, int n_in,
                              void* d_out, int out_size, void* d_ws, size_t ws_size,
                              hipStream_t stream) {
  const float* X  = (const float*)d_in[0];
  const float* W1 = (const float*)d_in[1];
  const float* B1 = (const float*)d_in[2];
  const float* W2 = (const float*)d_in[3];
  const float* B2 = (const float*)d_in[4];
  const int*   snd = (const int*)d_in[5];
  const int*   rcv = (const int*)d_in[6];
  float* out = (float*)d_out;
  char*  ws  = (char*)d_ws;

  float*    CADJ   = (float*)(ws + WS_CADJ_BYTES);
  unsigned* keep   = (ws_size >= WS_MIN_BYTES)   ? (unsigned*)(ws + WS_KEEP_BYTES)   : nullptr;
  unsigned* bitmap = (ws_size >= WS_DEDUP_BYTES) ? (unsigned*)(ws + WS_BITMAP_BYTES) : nullptr;
  const long long bm_words = bitmap ? (long long)N_NODES * N_NODES / 32 : 0;

  zero_kernel<<<1024, 256, 0, stream>>>(out + OUT_COARSE_NODES, CADJ, bitmap, bm_words);
  mlp_softmax_kernel<<<N_NODES / 64, 128, 0, stream>>>(X, W1, B1, W2, B2, out + OUT_ASSIGN);
  if (keep)
    keep_kernel<<<N_EDGES / 256, 256, 0, stream>>>(snd, rcv, keep, bitmap);
  coarse_nodes_kernel<<<1024, 128, 0, stream>>>(out + OUT_ASSIGN, X, out + OUT_COARSE_NODES);
  coarse_adj_kernel<<<(256 * 16) / 4, 128, 0, stream>>>(out + OUT_ASSIGN, snd, rcv, keep, CADJ);
  finalize_kernel<<<1, 32, 0, stream>>>(CADJ, out);
}